// randomWalk_86139864089342
// MI455X (gfx1250) — compile-verified
//
#include <hip/hip_runtime.h>
#include <hip/hip_bf16.h>

// ---- constants mirroring the reference ----
#define N_A      3550
#define N_D      20000
#define DUMMY    120000
#define START_D  9100
#define LWALK    20
#define CTX      7
#define WPN      100
#define BATCHN   2048
#define WTOT     (BATCHN * WPN)            // 204800 walks
#define NUM_WIN  (1 + LWALK + 1 - CTX)     // 15 context windows
#define POS_TOTAL (NUM_WIN * WTOT * CTX)   // 21,504,000 int32

typedef __attribute__((ext_vector_type(4))) unsigned int v4u;
typedef __attribute__((ext_vector_type(8))) int          v8i;
typedef __attribute__((ext_vector_type(4))) int          v4i;

// Stage a 1-D row of `nelem` int32 from global into LDS with the Tensor Data
// Mover (TDM).  D# layout per CDNA5 ISA §8.3/8.4:
//   group0: count=1, lds_addr, global_addr[56:0], type=2
//   group1: data_size=4B, tensor_dim0=nelem, tensor_dim1=1,
//           tile_dim0=nelem (<=65535), tile_dim1=1, stride0=nelem
//   groups 2/3: zero (tile <= 2D)
// Toolchain is the clang-23 6-arg builtin form:
//   (v4u g0, v8i g1, v4i, v4i, v8i, i32 cpol)
__device__ __forceinline__ void tdm_load_row_to_lds(const void* gaddr,
                                                    unsigned lds_byte_off,
                                                    unsigned nelem) {
  unsigned long long ga = (unsigned long long)gaddr;
  v4u g0;
  g0.s0 = 1u;                                              // count=1, user mode
  g0.s1 = lds_byte_off;                                    // lds_addr
  g0.s2 = (unsigned)ga;                                    // global_addr[31:0]
  g0.s3 = (unsigned)((ga >> 32) & 0x01FFFFFFu) | (2u << 30); // addr[56:32], type=2
  v8i g1;
  g1.s0 = (int)(2u << 16);                                 // data_size = 4 bytes
  g1.s1 = (int)((nelem & 0xFFFFu) << 16);                  // tensor_dim0[15:0]
  g1.s2 = (int)(((nelem >> 16) & 0xFFFFu) | (1u << 16));   // dim0[31:16], tensor_dim1=1
  g1.s3 = (int)((nelem & 0xFFFFu) << 16);                  // tile_dim0 = nelem
  g1.s4 = 1;                                               // tile_dim1=1, tile_dim2=0
  g1.s5 = (int)nelem;                                      // tensor_dim0_stride lo
  g1.s6 = 0;                                               // stride0 hi / stride1 lo
  g1.s7 = 0;
  v4i z4 = (v4i)0;
  v8i z8 = (v8i)0;
  __builtin_amdgcn_tensor_load_to_lds(g0, g1, z4, z4, z8, 0);
}

__device__ __forceinline__ int remap_node(int v) {
  // mask2: (v != DUMMY) && (v >= N_A)  ->  v - N_A + START_D
  return (v != DUMMY && v >= N_A) ? (v + (START_D - N_A)) : v;
}

// ---- positive walks: CSR neighbor sampling, rowptrs staged in LDS via TDM ----
__global__ void pos_walk_kernel(const float* __restrict__ pos_rand,
                                const int*   __restrict__ batch,
                                const int*   __restrict__ rowptr_ad,
                                const int*   __restrict__ col_ad,
                                const int*   __restrict__ rowptr_da,
                                const int*   __restrict__ col_da,
                                int col_ad_len, int col_da_len,
                                int* __restrict__ out) {
  extern __shared__ int lds[];
  int* l_rp_ad = lds;          // 3551 ints @ LDS byte 0      (14204 B)
  int* l_rp_da = lds + 3552;   // 20001 ints @ LDS byte 14208 (80004 B)

  if (threadIdx.x == 0) {
    tdm_load_row_to_lds(rowptr_ad, 0u,     (unsigned)(N_A + 1));
    tdm_load_row_to_lds(rowptr_da, 14208u, (unsigned)(N_D + 1));
    __builtin_amdgcn_s_wait_tensorcnt(0);  // TDM done -> LDS valid
  }
  __syncthreads();

  const int stride = gridDim.x * blockDim.x;
  for (int w = blockIdx.x * blockDim.x + threadIdx.x; w < WTOT; w += stride) {
    // vector-load this walk's 20 uniforms (rows are 80 B, 16-B aligned)
    float u[LWALK];
    const float4* pr = (const float4*)(pos_rand + (size_t)w * LWALK);
#pragma unroll
    for (int k = 0; k < LWALK / 4; ++k) {
      float4 t = pr[k];
      u[4 * k + 0] = t.x; u[4 * k + 1] = t.y;
      u[4 * k + 2] = t.z; u[4 * k + 3] = t.w;
    }

    int rw[LWALK + 1];
    int cur = batch[w & (BATCHN - 1)];   // jnp.tile(batch, WPN)
    rw[0] = cur;                         // position 0: a-type, offset 0

#pragma unroll
    for (int i = 0; i < LWALK; ++i) {
      const int* rp   = (i & 1) ? l_rp_da  : l_rp_ad;   // LDS gather
      const int* cl   = (i & 1) ? col_da   : col_ad;    // global gather (L2)
      const int  clen = (i & 1) ? col_da_len : col_ad_len;

      const bool alive = cur < DUMMY;
      const int  safe  = alive ? cur : 0;
      const int  lo    = rp[safe];
      const int  hi    = rp[safe + 1];
      const int  rc    = alive ? (hi - lo) : 0;
      const bool ok    = rc > 0;
      const int  off   = ok ? lo : 0;
      int idx = (int)(u[i] * (float)rc) + off;          // trunc, matches astype(i32)
      idx = min(max(idx, 0), clen - 1);                 // jnp.clip
      const int nxt = cl[idx];
      cur = ok ? nxt : DUMMY;

      // position i+1 is odd when i is even -> +N_A, then clip (> DUMMY -> DUMMY)
      int v = cur + ((i & 1) ? 0 : N_A);
      rw[i + 1] = (v > DUMMY) ? DUMMY : v;
    }

    // windows: out[(j*W + w)*C + c] = remap(rw[j + c])
#pragma unroll
    for (int j = 0; j < NUM_WIN; ++j) {
      const int base = (j * WTOT + w) * CTX;
#pragma unroll
      for (int c = 0; c < CTX; ++c)
        out[base + c] = remap_node(rw[j + c]);
    }
  }
}

// ---- negative walks: pure ALU resampling, one thread per walk ----
__global__ void neg_walk_kernel(const float* __restrict__ neg_rand,
                                const int*   __restrict__ batch,
                                int* __restrict__ out) {
  const int w = blockIdx.x * blockDim.x + threadIdx.x;
  if (w >= WTOT) return;

  float u[LWALK];
  const float4* nr = (const float4*)(neg_rand + (size_t)w * LWALK);
#pragma unroll
  for (int k = 0; k < LWALK / 4; ++k) {
    float4 t = nr[k];
    u[4 * k + 0] = t.x; u[4 * k + 1] = t.y;
    u[4 * k + 2] = t.z; u[4 * k + 3] = t.w;
  }

  int rw[LWALK + 1];
  rw[0] = batch[w & (BATCHN - 1)];
#pragma unroll
  for (int i = 0; i < LWALK; ++i) {
    const int n   = (i & 1) ? N_A : N_D;        // target type of metapath step
    const int cur = (int)(u[i] * (float)n);     // trunc toward zero
    rw[i + 1] = cur + ((i & 1) ? 0 : N_A);      // OFFSET on odd positions
  }

#pragma unroll
  for (int j = 0; j < NUM_WIN; ++j) {
    const int base = (j * WTOT + w) * CTX;
#pragma unroll
    for (int c = 0; c < CTX; ++c)
      out[base + c] = remap_node(rw[j + c]);
  }
}

extern "C" void kernel_launch(void* const* d_in, const int* in_sizes, int n_in,
                              void* d_out, int out_size, void* d_ws, size_t ws_size,
                              hipStream_t stream) {
  (void)n_in; (void)out_size; (void)d_ws; (void)ws_size;
  const float* pos_rand  = (const float*)d_in[0];
  const float* neg_rand  = (const float*)d_in[1];
  const int*   batch     = (const int*)d_in[2];
  const int*   rowptr_ad = (const int*)d_in[3];
  const int*   col_ad    = (const int*)d_in[4];
  const int*   rowptr_da = (const int*)d_in[5];
  const int*   col_da    = (const int*)d_in[6];
  const int col_ad_len = in_sizes[4];
  const int col_da_len = in_sizes[6];
  int* out = (int*)d_out;   // reference output is int32 (4-byte elements)

  // LDS: 3552 ints (rowptr_ad region, padded) + 20001 ints (rowptr_da)
  const size_t smem = (size_t)(3552 + N_D + 1) * sizeof(int);  // 94,212 B < 320 KB/WGP

  // 400 blocks x 256 threads (8 wave32s/block); 94 KB LDS -> 3 blocks/WGP, 24 waves/WGP
  pos_walk_kernel<<<400, 256, smem, stream>>>(pos_rand, batch, rowptr_ad, col_ad,
                                              rowptr_da, col_da,
                                              col_ad_len, col_da_len, out);

  neg_walk_kernel<<<WTOT / 256, 256, 0, stream>>>(neg_rand, batch, out + POS_TOTAL);
}